// DPhysics_45037027066414
// MI455X (gfx1250) — compile-verified
//
#include <hip/hip_runtime.h>
#include <stdint.h>

#define HGT 128
#define WID 128

// sum over the 16 "point" lanes of each half-wave (both halves hold identical
// per-point values, so xor masks 1/2/4/8 give the full 16-point sum in every lane)
__device__ __forceinline__ float warp_sum16(float v) {
    v += __shfl_xor(v, 1, 32);
    v += __shfl_xor(v, 2, 32);
    v += __shfl_xor(v, 4, 32);
    v += __shfl_xor(v, 8, 32);
    return v;
}

// RK4 with constant derivative (reference op order; /6 folded to a multiply)
__device__ __forceinline__ float rk4f(float x, float xd) {
    const float dt = 0.01f;
    float k1 = dt * xd;
    float k2 = dt * (xd + 0.5f * k1);
    float k3 = dt * (xd + 0.5f * k2);
    float k4 = dt * (xd + k3);
    return x + (k1 + 2.f * k2 + 2.f * k3 + k4) * (1.0f / 6.0f);
}

// native CDNA5 V_TANH_F32 when the toolchain exposes it; ocml tanhf otherwise
__device__ __forceinline__ float fast_tanh(float x) {
#if __has_builtin(__builtin_amdgcn_tanhf)
    return __builtin_amdgcn_tanhf(x);
#elif __has_builtin(__builtin_amdgcn_tanh_f32)
    return __builtin_amdgcn_tanh_f32(x);
#else
    return tanhf(x);
#endif
}

// bilinear interp on the LDS-resident 128x128 height field (1/0.1 == 10 exact)
__device__ __forceinline__ float interp_lds(const float* g, float xq, float yq) {
    float xi = (xq + 6.4f) * 10.0f;
    xi = fminf(fmaxf(xi, 0.f), 126.f);
    float yi = (yq + 6.4f) * 10.0f;
    yi = fminf(fmaxf(yi, 0.f), 126.f);
    float fx = floorf(xi), fy = floorf(yi);
    int x0 = (int)fx, y0 = (int)fy;
    float wx = xi - fx, wy = yi - fy;
    const float* p = g + x0 * WID + y0;
    float z00 = p[0], z01 = p[1], z10 = p[WID], z11 = p[WID + 1];
    return z00 * (1.f - wx) * (1.f - wy) + z10 * wx * (1.f - wy)
         + z01 * (1.f - wx) * wy        + z11 * wx * wy;
}

extern "C" __global__ __launch_bounds__(32)
void dphys_sim_kernel(const float* __restrict__ z_grid,
                      const float* __restrict__ controls,
                      const float* __restrict__ robot_points,
                      const float* __restrict__ I_inv_g,
                      float* __restrict__ out, int T)
{
    extern __shared__ float tile[];            // 128*128 floats = 64 KB
    const int  b       = blockIdx.x;           // batch element
    const int  tid     = threadIdx.x;          // 0..31 (wave32)
    const int  p       = tid & 15;             // contact point id
    const bool hi_half = tid >= 16;

    // ---- CDNA5 async DMA: this batch's 64 KB height field -> LDS (ASYNCcnt).
    // Fully unrolled (128 straight-line issues, 512 B each); hardware stalls
    // issue if ASYNCcnt would overflow, so no manual throttling needed.
    {
        uint64_t gbase = (uint64_t)(uintptr_t)(z_grid + (size_t)b * (HGT * WID));
        unsigned lds0  = (unsigned)(uintptr_t)(&tile[0]);  // flat->LDS: low 32 bits
        unsigned off   = (unsigned)(tid * 16);
        #pragma unroll
        for (int it = 0; it < (HGT * WID * 4) / (32 * 16); ++it) {
            unsigned la = lds0 + off;
            asm volatile("global_load_async_to_lds_b128 %0, %1, %2"
                         :: "v"(la), "v"(off), "s"(gbase) : "memory");
            off += 32 * 16;
        }
    }

    // ---- per-lane constants while the DMA is in flight ----
    float ppx = robot_points[3 * p + 0];
    float ppy = robot_points[3 * p + 1];
    float ppz = robot_points[3 * p + 2];
    float cogy = warp_sum16(ppy) * 0.0625f;     // centroid y (matches pts.mean(0)[1])
    const bool ml = ppy > cogy;
    const bool mr = ppy < cogy;
    float Ii[9];
    #pragma unroll
    for (int i = 0; i < 9; ++i) Ii[i] = I_inv_g[i];

    // ---- initial state (replicated across lanes) ----
    float X[3]  = {0.f, 0.f, 1.f};
    float Xd[3] = {0.f, 0.f, 0.f};
    float Om[3] = {0.f, 0.f, 0.f};
    float R[9]  = {1.f, 0.f, 0.f,  0.f, 1.f, 0.f,  0.f, 0.f, 1.f};
    float xp[3] = {ppx, ppy, 1.f + ppz};        // x + robot_points
    float xdp[3]= {0.f, 0.f, 0.f};              // carried xd_points

    asm volatile("s_wait_asynccnt 0x0" ::: "memory");   // grid now in LDS

    const float K_STIFF = 5000.f;
    const float K_DAMP  = 894.4271909999159f;   // sqrt(4*40*5000)
    const float EPS     = 1e-6f;
    const float DT      = 0.01f;

    for (int t = 0; t < T; ++t) {
        float u_l = controls[((size_t)b * T + t) * 2 + 0];
        float u_r = controls[((size_t)b * T + t) * 2 + 1];

        // --- height + stencil: halves split the 4 derivative taps (3 interps/lane) ---
        float i_z = interp_lds(tile, xp[0], xp[1]);
        float dx1 = hi_half ? 0.f  : 0.1f;
        float dy1 = hi_half ? 0.1f : 0.f;
        float i_p = interp_lds(tile, xp[0] + dx1, xp[1] + dy1);
        float i_m = interp_lds(tile, xp[0] - dx1, xp[1] - dy1);
        float num_own   = i_p - i_m;
        float num_other = __shfl_xor(num_own, 16, 32);
        float dzdx = (hi_half ? num_other : num_own)   * 5.0f;   // /(2*0.1)
        float dzdy = (hi_half ? num_own   : num_other) * 5.0f;

        // --- contact + normal ---
        float dh = xp[2] - i_z;
        bool  on_grid = (xp[0] >= -6.4f) && (xp[0] <= 6.4f) &&
                        (xp[1] >= -6.4f) && (xp[1] <= 6.4f);
        float contact = ((dh <= 0.f) && on_grid) ? 1.f : 0.f;
        float nx = -dzdx, ny = -dzdy, nz = 1.f;
        float ninv = 1.0f / (sqrtf(nx * nx + ny * ny + nz * nz) + EPS);
        nx *= ninv; ny *= ninv; nz *= ninv;

        // --- spring force ---
        float xdn = xdp[0] * nx + xdp[1] * ny + xdp[2] * nz;
        float mag = -(K_STIFF * dh + K_DAMP * xdn) * contact;
        float Fs[3] = {mag * nx, mag * ny, mag * nz};

        // --- friction (thrust = normalized first column of R) ---
        float tx = R[0], ty = R[3], tz = R[6];
        float tinv = 1.0f / (sqrtf(tx * tx + ty * ty + tz * tz) + EPS);
        tx *= tinv; ty *= tinv; tz *= tinv;
        float Nmag  = sqrtf(Fs[0] * Fs[0] + Fs[1] * Fs[1] + Fs[2] * Fs[2]);
        float uv    = ml ? u_l : u_r;
        float fscal = (ml || mr) ? (0.5f * Nmag) : 0.f;
        float Ff[3];
        Ff[0] = fscal * fast_tanh(uv * tx - xdp[0]);
        Ff[1] = fscal * fast_tanh(uv * ty - xdp[1]);
        Ff[2] = fscal * fast_tanh(uv * tz - xdp[2]);

        // --- torque and force sums over the 16 points (wave reductions) ---
        float rx = xp[0] - X[0], ry = xp[1] - X[1], rz = xp[2] - X[2];
        float Ftx = Fs[0] + Ff[0], Fty = Fs[1] + Ff[1], Ftz = Fs[2] + Ff[2];
        float Tq[3] = { warp_sum16(ry * Ftz - rz * Fty),
                        warp_sum16(rz * Ftx - rx * Ftz),
                        warp_sum16(rx * Fty - ry * Ftx) };
        float Ssum[3] = { warp_sum16(Fs[0]), warp_sum16(Fs[1]), warp_sum16(Fs[2]) };
        float Fsum[3] = { warp_sum16(Ff[0]), warp_sum16(Ff[1]), warp_sum16(Ff[2]) };

        float omd[3];
        #pragma unroll
        for (int i = 0; i < 3; ++i)
            omd[i] = Ii[i * 3 + 0] * Tq[0] + Ii[i * 3 + 1] * Tq[1] + Ii[i * 3 + 2] * Tq[2];

        float Fc[3] = { Ssum[0] * 0.0625f + Fsum[0] * 0.0625f,
                        Ssum[1] * 0.0625f + Fsum[1] * 0.0625f,
                        -40.f * 9.81f + Ssum[2] * 0.0625f + Fsum[2] * 0.0625f };
        float xdd[3] = { Fc[0] * 0.025f, Fc[1] * 0.025f, Fc[2] * 0.025f };  // /40

        // xd_points_new = xd + omega x r   (old xd, old omega)
        float xdpn[3] = { Xd[0] + (Om[1] * rz - Om[2] * ry),
                          Xd[1] + (Om[2] * rx - Om[0] * rz),
                          Xd[2] + (Om[0] * ry - Om[1] * rx) };

        // --- integrate (reference order: xd, then x with xd_new, points, omega) ---
        float Xdn[3], Xn[3], xpn[3], Omn[3];
        #pragma unroll
        for (int c = 0; c < 3; ++c) {
            Xdn[c] = rk4f(Xd[c], xdd[c]);
            Xn[c]  = rk4f(X[c], Xdn[c]);
            xpn[c] = rk4f(xp[c], xdpn[c]);
            Omn[c] = rk4f(Om[c], omd[c]);
        }

        // --- rotation update: R_new = R (I + A sin(th dt) + A^2 (1-cos(th dt))) ---
        float th   = sqrtf(Omn[0] * Omn[0] + Omn[1] * Omn[1] + Omn[2] * Omn[2]);
        float dinv = 1.0f / (th + EPS);
        float ax = Omn[0] * dinv, ay = Omn[1] * dinv, az = Omn[2] * dinv;
        float A[9]  = {0.f, -az, ay,   az, 0.f, -ax,   -ay, ax, 0.f};
        float A2[9];
        #pragma unroll
        for (int i = 0; i < 3; ++i)
            #pragma unroll
            for (int j = 0; j < 3; ++j)
                A2[i * 3 + j] = A[i * 3 + 0] * A[0 + j] + A[i * 3 + 1] * A[3 + j]
                              + A[i * 3 + 2] * A[6 + j];
        float st = sinf(th * DT), ct = 1.f - cosf(th * DT);
        float M[9];
        #pragma unroll
        for (int i = 0; i < 9; ++i) M[i] = A[i] * st + A2[i] * ct;
        M[0] += 1.f; M[4] += 1.f; M[8] += 1.f;
        float Rn[9];
        #pragma unroll
        for (int i = 0; i < 3; ++i)
            #pragma unroll
            for (int j = 0; j < 3; ++j)
                Rn[i * 3 + j] = R[i * 3 + 0] * M[0 + j] + R[i * 3 + 1] * M[3 + j]
                              + R[i * 3 + 2] * M[6 + j];

        // --- emit traj[b, t, 162] = [x(3) xd(3) om(3) R(9) xpts(48) Fs(48) Ff(48)] ---
        size_t base = ((size_t)b * (size_t)T + (size_t)t) * 162;
        if (tid == 0) {
            #pragma unroll
            for (int c = 0; c < 3; ++c) {
                out[base + c]     = Xn[c];
                out[base + 3 + c] = Xdn[c];
                out[base + 6 + c] = Omn[c];
            }
            #pragma unroll
            for (int i = 0; i < 9; ++i) out[base + 9 + i] = Rn[i];
        }
        if (tid < 16) {
            #pragma unroll
            for (int c = 0; c < 3; ++c) {
                out[base + 18  + 3 * p + c] = xpn[c];
                out[base + 66  + 3 * p + c] = Fs[c];
                out[base + 114 + 3 * p + c] = Ff[c];
            }
        }

        // --- carry ---
        #pragma unroll
        for (int c = 0; c < 3; ++c) {
            X[c] = Xn[c]; Xd[c] = Xdn[c]; Om[c] = Omn[c];
            xp[c] = xpn[c]; xdp[c] = xdpn[c];
        }
        #pragma unroll
        for (int i = 0; i < 9; ++i) R[i] = Rn[i];
    }
}

extern "C" void kernel_launch(void* const* d_in, const int* in_sizes, int n_in,
                              void* d_out, int out_size, void* d_ws, size_t ws_size,
                              hipStream_t stream) {
    (void)n_in; (void)out_size; (void)d_ws; (void)ws_size;
    const float* z_grid       = (const float*)d_in[0];
    const float* controls     = (const float*)d_in[1];
    const float* robot_points = (const float*)d_in[2];
    const float* I_inv        = (const float*)d_in[3];
    // masks (d_in[4], d_in[5]) are re-derived from robot_points geometry on-device

    int B = in_sizes[0] / (HGT * WID);
    int T = in_sizes[1] / (B * 2);

    dphys_sim_kernel<<<B, 32, HGT * WID * sizeof(float), stream>>>(
        z_grid, controls, robot_points, I_inv, (float*)d_out, T);
}